// BigChunk1_76579266888101
// MI455X (gfx1250) — compile-verified
//
#include <hip/hip_runtime.h>
#include <hip/hip_bf16.h>

typedef float v2f __attribute__((ext_vector_type(2)));
typedef float v8f __attribute__((ext_vector_type(8)));

#define HDIM 2048
#define NHEAD 8
#define HD 256
#define FFND 4096
#define PLD 256
#define LTOT 30
#define NLAY 15
#define WSL 512
#define CTXL 1024
#define POSI 100
#define EPSF 1e-6f
#define NKEY 112   // 7 tiles of 16 cover keys 0..100 (mask kills the rest)

// ---------------- device helpers ----------------

__device__ __forceinline__ float block_sum256(float v, float* red) {
  int t = threadIdx.x;
  red[t] = v; __syncthreads();
#pragma unroll
  for (int s = 128; s > 0; s >>= 1) { if (t < s) red[t] += red[t + s]; __syncthreads(); }
  float r = red[0]; __syncthreads();
  return r;
}

__device__ __forceinline__ float wave_sum(float v) {
#pragma unroll
  for (int o = 16; o > 0; o >>= 1) v += __shfl_xor(v, o, 32);
  return v;
}
__device__ __forceinline__ float wave_max(float v) {
#pragma unroll
  for (int o = 16; o > 0; o >>= 1) v = fmaxf(v, __shfl_xor(v, o, 32));
  return v;
}

__device__ __forceinline__ float gelu_tanh(float x) {
  const float c0 = 0.7978845608028654f, c1 = 0.044715f;
  return 0.5f * x * (1.f + tanhf(c0 * (x + c1 * x * x * x)));
}

__device__ __forceinline__ v8f wmma_f32_4(v2f a, v2f b, v8f c) {
#if __has_builtin(__builtin_amdgcn_wmma_f32_16x16x4_f32)
  return __builtin_amdgcn_wmma_f32_16x16x4_f32(false, a, false, b, (short)0, c, false, false);
#else
  return c;
#endif
}

// ---------------- kernels ----------------

// PLE projection: proj[p] = dot(h0, Wc[p,:]) * H^-0.5   (7680 rows, wave per row)
__global__ void ple_proj_kernel(const float* __restrict__ h0,
                                const float* __restrict__ Wc,
                                float* __restrict__ proj) {
  int t = threadIdx.x, wave = t >> 5, lane = t & 31;
  int row = blockIdx.x * 8 + wave;
  const float* r = Wc + (size_t)row * HDIM;
  float acc = 0.f;
  for (int i = lane; i < HDIM; i += 32) acc += h0[i] * r[i];
  acc = wave_sum(acc);
  if (lane == 0) proj[row] = acc * 0.022097086912079608f; // 2048^-0.5
}

// PLE group norm: one block per group of 256
__global__ void ple_norm_kernel(const float* __restrict__ proj,
                                const float* __restrict__ normw,
                                const float* __restrict__ plr,
                                float* __restrict__ pleout) {
  __shared__ float red[256];
  int g = blockIdx.x, t = threadIdx.x;
  float x = proj[g * PLD + t];
  float ss = block_sum256(x * x, red);
  float rs = rsqrtf(ss / (float)PLD + EPSF);
  pleout[g * PLD + t] = (x * rs * normw[t] + plr[g * PLD + t]) * 0.70710678118654752f;
}

// QKV: hn = rms(h, w_in); q = hn@Wq (2048), k = hn@Wk (256), v = hn@Wv (256)
// grid = 40 blocks x 256 threads; block: 64 outputs x 4 h-partitions
__global__ void qkv_kernel(const float* __restrict__ h, const float* __restrict__ w_in_l,
                           const float* __restrict__ Wq, const float* __restrict__ Wk,
                           const float* __restrict__ Wv,
                           float* __restrict__ q, float* __restrict__ kvec,
                           float* __restrict__ vvec) {
  __shared__ float hn[HDIM];
  __shared__ float red[256];
  int t = threadIdx.x;
  float ss = 0.f;
  for (int i = t; i < HDIM; i += 256) { float x = h[i]; hn[i] = x; ss += x * x; }
  ss = block_sum256(ss, red);
  float rs = rsqrtf(ss / (float)HDIM + EPSF);
  for (int i = t; i < HDIM; i += 256) hn[i] = hn[i] * rs * (1.f + w_in_l[i]);
  __syncthreads();

  int b = blockIdx.x;
  const float* W; float* out; int O, obase;
  if (b < 32)      { W = Wq; out = q;    O = 2048; obase = b * 64; }
  else if (b < 36) { W = Wk; out = kvec; O = 256;  obase = (b - 32) * 64; }
  else             { W = Wv; out = vvec; O = 256;  obase = (b - 36) * 64; }

  int o = obase + (t & 63), part = t >> 6;
  float acc = 0.f;
  int ibeg = part * 512, iend = ibeg + 512;
  for (int i0 = ibeg; i0 < iend; i0 += 8) {
    if (i0 + 64 < iend) __builtin_prefetch(&W[(size_t)(i0 + 64) * O + o], 0, 0);
#pragma unroll
    for (int i = i0; i < i0 + 8; ++i) acc += hn[i] * W[(size_t)i * O + o];
  }
  red[t] = acc; __syncthreads();
  if (t < 64) out[obase + t] = red[t] + red[t + 64] + red[t + 128] + red[t + 192];
}

// q/k RMSNorm + RoPE; write roped q back, write roped k / v into cache row POS
__global__ void ropekv_kernel(float* __restrict__ q, const float* __restrict__ kvec,
                              const float* __restrict__ vvec,
                              const float* __restrict__ wqn, const float* __restrict__ wkn,
                              const float* __restrict__ cosv, const float* __restrict__ sinv,
                              float* __restrict__ Kdst, float* __restrict__ Vdst) {
  __shared__ float qn[NHEAD * HD];
  __shared__ float kn[HD];
  __shared__ float red[256];
  int t = threadIdx.x, head = t >> 5, lane = t & 31;

  float ss = 0.f;
  for (int d = lane; d < HD; d += 32) { float x = q[head * HD + d]; ss += x * x; }
  ss = wave_sum(ss);
  float rs = rsqrtf(ss / (float)HD + EPSF);
  for (int d = lane; d < HD; d += 32) qn[head * HD + d] = q[head * HD + d] * rs * (1.f + wqn[d]);

  float kx = kvec[t];
  float ks = block_sum256(kx * kx, red); // barrier also covers qn writes
  float krs = rsqrtf(ks / (float)HD + EPSF);
  kn[t] = kx * krs * (1.f + wkn[t]);
  __syncthreads();

  for (int idx = t; idx < NHEAD * HD; idx += 256) {
    int d = idx & 255;
    float x = qn[idx];
    float rot = (d < 128) ? -qn[idx + 128] : qn[idx - 128];
    q[idx] = x * cosv[d] + rot * sinv[d];
  }
  {
    int d = t;
    float x = kn[d];
    float rot = (d < 128) ? -kn[d + 128] : kn[d - 128];
    Kdst[POSI * HD + d] = x * cosv[d] + rot * sinv[d];
    Vdst[POSI * HD + d] = vvec[d];
  }
}

// Attention via V_WMMA_F32_16X16X4_F32. One block, 8 waves.
// Q and att are staged in LDS zero-padded to M=16 so A-fragment reads are
// unpredicated ds_load_b64 (no EXEC divergence inside the wmma loops).
__global__ void attn_kernel(const float* __restrict__ qbuf, const float* __restrict__ Kc,
                            const float* __restrict__ Vc, const float* __restrict__ mask,
                            float* __restrict__ ctx) {
  __shared__ float qs[16 * HD];       // rows 8..15 zero
  __shared__ float attL[16 * NKEY];   // rows 8..15 zero
  int t = threadIdx.x, wave = t >> 5, lane = t & 31;
  int hi = lane >> 4, m16 = lane & 15;
  const float scale = 0.0625f; // 256^-0.5

  for (int idx = t; idx < 16 * HD; idx += 256)
    qs[idx] = (idx < NHEAD * HD) ? qbuf[idx] : 0.f;
  for (int idx = t; idx < 8 * NKEY; idx += 256)
    attL[8 * NKEY + idx] = 0.f;
  __syncthreads();

  // Phase 1: scores = Q x K^T  (M=16, N=112 in 7 tiles, K=256)
  if (wave < NKEY / 16) {
    int nt = wave;
    const float* krow = Kc + (size_t)(nt * 16 + m16) * HD;
    v8f c = {};
#pragma unroll 4
    for (int kb = 0; kb < HD; kb += 4) {
      int ka = kb + 2 * hi;
      v2f a, b;
      a.x = qs[m16 * HD + ka];
      a.y = qs[m16 * HD + ka + 1];
      b.x = krow[ka];
      b.y = krow[ka + 1];
      c = wmma_f32_4(a, b, c);
    }
    if (lane < 16) {
#pragma unroll
      for (int j = 0; j < 8; ++j)
        attL[j * NKEY + nt * 16 + lane] = c[j] * scale + mask[nt * 16 + lane];
    }
  }
  __syncthreads();

  // Phase 2: softmax, wave per head (keys 101..111 carry -1e9 mask -> exp==0)
  {
    int h8 = wave;
    float vals[4], mx = -3.0e38f;
#pragma unroll
    for (int i = 0; i < 4; ++i) {
      int key = lane + 32 * i;
      vals[i] = (key < NKEY) ? attL[h8 * NKEY + key] : -3.0e38f;
      mx = fmaxf(mx, vals[i]);
    }
    mx = wave_max(mx);
    float s = 0.f;
#pragma unroll
    for (int i = 0; i < 4; ++i) {
      int key = lane + 32 * i;
      if (key < NKEY) { float e = __expf(vals[i] - mx); attL[h8 * NKEY + key] = e; s += e; }
    }
    s = wave_sum(s);
    float inv = 1.f / s;
#pragma unroll
    for (int i = 0; i < 4; ++i) {
      int key = lane + 32 * i;
      if (key < NKEY) attL[h8 * NKEY + key] *= inv;
    }
  }
  __syncthreads();

  // Phase 3: ctx = att @ V ; M=16, N=256 (16 tiles, 2 per wave), K=112
  for (int r = 0; r < 2; ++r) {
    int nt = wave * 2 + r;
    v8f c = {};
#pragma unroll 4
    for (int kb = 0; kb < NKEY; kb += 4) {
      int ka = kb + 2 * hi;
      v2f a, b;
      a.x = attL[m16 * NKEY + ka];
      a.y = attL[m16 * NKEY + ka + 1];
      b.x = Vc[(size_t)ka * HD + nt * 16 + m16];
      b.y = Vc[(size_t)(ka + 1) * HD + nt * 16 + m16];
      c = wmma_f32_4(a, b, c);
    }
    if (lane < 16) {
#pragma unroll
      for (int j = 0; j < 8; ++j)
        ctx[j * HD + nt * 16 + lane] = c[j];
    }
  }
}

// h[o] += dot(ctx, Wo[:,o]) ; 32 blocks x (64 outputs x 4 partitions)
__global__ void wo_kernel(const float* __restrict__ ctxv, const float* __restrict__ W,
                          float* __restrict__ hio) {
  __shared__ float xs[HDIM];
  __shared__ float red[256];
  int t = threadIdx.x;
  for (int i = t; i < HDIM; i += 256) xs[i] = ctxv[i];
  __syncthreads();
  int obase = blockIdx.x * 64, o = t & 63, part = t >> 6;
  float acc = 0.f;
  int ibeg = part * 512, iend = ibeg + 512;
  for (int i0 = ibeg; i0 < iend; i0 += 8) {
    if (i0 + 64 < iend) __builtin_prefetch(&W[(size_t)(i0 + 64) * HDIM + obase + o], 0, 0);
#pragma unroll
    for (int i = i0; i < i0 + 8; ++i) acc += xs[i] * W[(size_t)i * HDIM + obase + o];
  }
  red[t] = acc; __syncthreads();
  if (t < 64) hio[obase + t] += red[t] + red[t + 64] + red[t + 128] + red[t + 192];
}

// FFN gate/up: hn2 = rms(h, w_post); t[o] = gelu(hn2@Wg) * (hn2@Wu) ; 64 blocks
__global__ void ffn_gu_kernel(const float* __restrict__ h, const float* __restrict__ w_post_l,
                              const float* __restrict__ Wg, const float* __restrict__ Wu,
                              float* __restrict__ tbuf) {
  __shared__ float hn2[HDIM];
  __shared__ float redg[256];
  __shared__ float redu[256];
  int t = threadIdx.x;
  float ss = 0.f;
  for (int i = t; i < HDIM; i += 256) { float x = h[i]; hn2[i] = x; ss += x * x; }
  ss = block_sum256(ss, redg);
  float rs = rsqrtf(ss / (float)HDIM + EPSF);
  for (int i = t; i < HDIM; i += 256) hn2[i] = hn2[i] * rs * (1.f + w_post_l[i]);
  __syncthreads();
  int obase = blockIdx.x * 64, o = t & 63, part = t >> 6;
  float ag = 0.f, au = 0.f;
  int ibeg = part * 512, iend = ibeg + 512;
  for (int i0 = ibeg; i0 < iend; i0 += 8) {
    if (i0 + 64 < iend) {
      __builtin_prefetch(&Wg[(size_t)(i0 + 64) * FFND + obase + o], 0, 0);
      __builtin_prefetch(&Wu[(size_t)(i0 + 64) * FFND + obase + o], 0, 0);
    }
#pragma unroll
    for (int i = i0; i < i0 + 8; ++i) {
      float x = hn2[i];
      ag += x * Wg[(size_t)i * FFND + obase + o];
      au += x * Wu[(size_t)i * FFND + obase + o];
    }
  }
  redg[t] = ag; redu[t] = au; __syncthreads();
  if (t < 64) {
    float g = redg[t] + redg[t + 64] + redg[t + 128] + redg[t + 192];
    float u = redu[t] + redu[t + 64] + redu[t + 128] + redu[t + 192];
    tbuf[obase + t] = gelu_tanh(g) * u;
  }
}

// h[o] += dot(t, Wd[:,o]) + dot(gelu(pli), W_ple[l][:,o]) ; 32 blocks
__global__ void ffn_down_ple_kernel(const float* __restrict__ tbuf, const float* __restrict__ Wd,
                                    const float* __restrict__ pleall, const float* __restrict__ Wple_l,
                                    float* __restrict__ hio, int l) {
  __shared__ float ts[FFND];
  __shared__ float gp[PLD];
  __shared__ float red[256];
  int t = threadIdx.x;
  for (int i = t; i < FFND; i += 256) ts[i] = tbuf[i];
  gp[t] = gelu_tanh(pleall[l * PLD + t]);
  __syncthreads();
  int obase = blockIdx.x * 64, o = t & 63, part = t >> 6;
  float acc = 0.f;
  int ibeg = part * 1024, iend = ibeg + 1024;
  for (int i0 = ibeg; i0 < iend; i0 += 8) {
    if (i0 + 64 < iend) __builtin_prefetch(&Wd[(size_t)(i0 + 64) * HDIM + obase + o], 0, 0);
#pragma unroll
    for (int i = i0; i < i0 + 8; ++i) acc += ts[i] * Wd[(size_t)i * HDIM + obase + o];
  }
  if (part == 0)
    for (int i = 0; i < PLD; ++i)
      acc += gp[i] * Wple_l[(size_t)i * HDIM + obase + o];
  red[t] = acc; __syncthreads();
  if (t < 64) hio[obase + t] += red[t] + red[t + 64] + red[t + 128] + red[t + 192];
}

// ---------------- host ----------------

extern "C" void kernel_launch(void* const* d_in, const int* in_sizes, int n_in,
                              void* d_out, int out_size, void* d_ws, size_t ws_size,
                              hipStream_t stream) {
  (void)in_sizes; (void)n_in; (void)out_size; (void)ws_size;
  const float* h0        = (const float*)d_in[0];
  const float* mask_full = (const float*)d_in[1];
  const float* mask_sld  = (const float*)d_in[2];
  const float* plr       = (const float*)d_in[4];
  const float* cos_s     = (const float*)d_in[5];
  const float* sin_s     = (const float*)d_in[6];
  const float* cos_f     = (const float*)d_in[7];
  const float* sin_f     = (const float*)d_in[8];
  const float* Ks_in     = (const float*)d_in[9];
  const float* Vs_in     = (const float*)d_in[10];
  const float* Kf_in     = (const float*)d_in[11];
  const float* Vf_in     = (const float*)d_in[12];
  const float* ple_conv  = (const float*)d_in[13];
  const float* ple_nw    = (const float*)d_in[14];
  const float* Wple      = (const float*)d_in[15];
  const float* Wq        = (const float*)d_in[16];
  const float* Wk        = (const float*)d_in[17];
  const float* Wv        = (const float*)d_in[18];
  const float* Wo        = (const float*)d_in[19];
  const float* w_in      = (const float*)d_in[20];
  const float* w_post    = (const float*)d_in[21];
  const float* w_qn      = (const float*)d_in[22];
  const float* w_kn      = (const float*)d_in[23];
  const float* Wg        = (const float*)d_in[24];
  const float* Wu        = (const float*)d_in[25];
  const float* Wd        = (const float*)d_in[26];
  float* out = (float*)d_out;
  float* ws  = (float*)d_ws;

  // workspace layout (floats)
  float* hbuf = ws;            // 2048
  float* qbuf = ws + 2048;     // 2048
  float* kbuf = ws + 4096;     // 256
  float* vbuf = ws + 4352;     // 256
  float* cbuf = ws + 4608;     // 2048 (ctx)
  float* tbuf = ws + 6656;     // 4096
  float* proj = ws + 10752;    // 7680

  // output layout (floats)
  const size_t SLAB_S = (size_t)WSL * HD;      // 131072
  const size_t SLAB_F = (size_t)CTXL * HD;     // 262144
  const size_t OUT_KS = 2048;
  const size_t OUT_VS = OUT_KS + 13 * SLAB_S;
  const size_t OUT_KF = OUT_VS + 13 * SLAB_S;
  const size_t OUT_VF = OUT_KF + 2 * SLAB_F;
  const size_t OUT_PLE = OUT_VF + 2 * SLAB_F;
  const size_t OUT_K13 = OUT_PLE + LTOT * PLD;
  const size_t OUT_V13 = OUT_K13 + SLAB_S;
  const size_t OUT_K14 = OUT_V13 + SLAB_S;
  const size_t OUT_V14 = OUT_K14 + SLAB_F;

  hipMemcpyAsync(hbuf, h0, HDIM * sizeof(float), hipMemcpyDeviceToDevice, stream);

  ple_proj_kernel<<<960, 256, 0, stream>>>(h0, ple_conv, proj);
  ple_norm_kernel<<<LTOT, 256, 0, stream>>>(proj, ple_nw, plr, out + OUT_PLE);

  int si = 0, fi = 0;
  for (int l = 0; l < NLAY; ++l) {
    bool is_full = (l == 4) || (l == 14);
    const float* cosv = is_full ? cos_f : cos_s;
    const float* sinv = is_full ? sin_f : sin_s;
    const float* mask = is_full ? mask_full : mask_sld;

    float* Kdst; float* Vdst; const float* Ksrc; const float* Vsrc; size_t slab;
    if (is_full) {
      slab = SLAB_F;
      Kdst = out + OUT_KF + fi * SLAB_F;  Ksrc = Kf_in + fi * SLAB_F;
      Vdst = out + OUT_VF + fi * SLAB_F;  Vsrc = Vf_in + fi * SLAB_F;
    } else {
      slab = SLAB_S;
      Kdst = out + OUT_KS + si * SLAB_S;  Ksrc = Ks_in + si * SLAB_S;
      Vdst = out + OUT_VS + si * SLAB_S;  Vsrc = Vs_in + si * SLAB_S;
    }
    hipMemcpyAsync(Kdst, Ksrc, slab * sizeof(float), hipMemcpyDeviceToDevice, stream);
    hipMemcpyAsync(Vdst, Vsrc, slab * sizeof(float), hipMemcpyDeviceToDevice, stream);

    qkv_kernel<<<40, 256, 0, stream>>>(hbuf, w_in + (size_t)l * HDIM,
                                       Wq + (size_t)l * HDIM * (NHEAD * HD),
                                       Wk + (size_t)l * HDIM * HD,
                                       Wv + (size_t)l * HDIM * HD,
                                       qbuf, kbuf, vbuf);
    ropekv_kernel<<<1, 256, 0, stream>>>(qbuf, kbuf, vbuf,
                                         w_qn + (size_t)l * HD, w_kn + (size_t)l * HD,
                                         cosv, sinv, Kdst, Vdst);
    attn_kernel<<<1, 256, 0, stream>>>(qbuf, Kdst, Vdst, mask, cbuf);
    wo_kernel<<<32, 256, 0, stream>>>(cbuf, Wo + (size_t)l * (NHEAD * HD) * HDIM, hbuf);
    ffn_gu_kernel<<<64, 256, 0, stream>>>(hbuf, w_post + (size_t)l * HDIM,
                                          Wg + (size_t)l * HDIM * FFND,
                                          Wu + (size_t)l * HDIM * FFND, tbuf);
    ffn_down_ple_kernel<<<32, 256, 0, stream>>>(tbuf, Wd + (size_t)l * FFND * HDIM,
                                                out + OUT_PLE,
                                                Wple + (size_t)l * PLD * HDIM, hbuf, l);

    if (l == 13) { // sliding slot 12 duplicate
      hipMemcpyAsync(out + OUT_K13, Kdst, SLAB_S * sizeof(float), hipMemcpyDeviceToDevice, stream);
      hipMemcpyAsync(out + OUT_V13, Vdst, SLAB_S * sizeof(float), hipMemcpyDeviceToDevice, stream);
    }
    if (l == 14) { // full slot 1 duplicate
      hipMemcpyAsync(out + OUT_K14, Kdst, SLAB_F * sizeof(float), hipMemcpyDeviceToDevice, stream);
      hipMemcpyAsync(out + OUT_V14, Vdst, SLAB_F * sizeof(float), hipMemcpyDeviceToDevice, stream);
    }
    if (is_full) ++fi; else ++si;
  }

  hipMemcpyAsync(out, hbuf, HDIM * sizeof(float), hipMemcpyDeviceToDevice, stream);
}